// SipHash_23287312679158
// MI455X (gfx1250) — compile-verified
//
#include <hip/hip_runtime.h>
#include <hip/hip_bf16.h>

// SipHash-2-4 over 4M rows of 16 fp32 (64 bytes) each -> int64 per row.
// CDNA5 (gfx1250, wave32) strategy:
//   * Stage each block's 256-row (16 KB) tile into LDS with
//     global_load_async_to_lds_b64 (ASYNCcnt path, CDNA5 ISA):
//     every instruction moves 256 contiguous bytes per wave -> perfect
//     coalescing of the 23.3 TB/s HBM stream.
//   * Full-tile fast path issues the 8 async copies unconditionally (rows is
//     an exact multiple of 256 here); a guarded tail path handles remainders
//     so the kernel stays correct for any row count.
//   * LDS rows padded 64B -> 72B (9 u64): lane t reads dwords starting at
//     t*18 mod 64; period 32 => all 32 lanes on distinct even bank pairs =>
//     conflict-free ds reads.
//   * SipHash state in u64 registers; rotates lower to 32-bit funnel shifts,
//     rotl-by-32 is a free half swap.

#define ROWS_PER_BLOCK 256
#define ROW_BYTES      64
#define LDS_ROW_U64    9                      // 72-byte padded LDS row
#define LDS_ROW_BYTES  (LDS_ROW_U64 * 8)      // 72
#define CHUNKS_PER_ROW 8                      // 8 x 8-byte async chunks per row
#define TILE_BYTES     (ROWS_PER_BLOCK * ROW_BYTES)        // 16384
#define K_GSTRIDE      (ROWS_PER_BLOCK * 8)                // 2048 B global per k-step
#define K_LSTRIDE      ((ROWS_PER_BLOCK / 8) * LDS_ROW_BYTES) // 2304 B LDS per k-step

__device__ __forceinline__ unsigned long long rotl64(unsigned long long x, unsigned b) {
    return (x << b) | (x >> (64u - b));
}

__device__ __forceinline__ void sipround(unsigned long long& v0, unsigned long long& v1,
                                         unsigned long long& v2, unsigned long long& v3) {
    v0 += v1; v1 = rotl64(v1, 13); v1 ^= v0; v0 = rotl64(v0, 32);
    v2 += v3; v3 = rotl64(v3, 16); v3 ^= v2;
    v0 += v3; v3 = rotl64(v3, 21); v3 ^= v0;
    v2 += v1; v1 = rotl64(v1, 17); v1 ^= v2; v2 = rotl64(v2, 32);
}

__global__ __launch_bounds__(ROWS_PER_BLOCK)
void siphash_rows_kernel(const float* __restrict__ x,
                         unsigned long long* __restrict__ out,
                         int rows) {
    __shared__ __align__(16) unsigned long long smem[ROWS_PER_BLOCK * LDS_ROW_U64];

    const int t = threadIdx.x;
    const long long tileRow0 = (long long)blockIdx.x * ROWS_PER_BLOCK;
    long long rem64 = (long long)rows - tileRow0;
    const int remaining = rem64 > ROWS_PER_BLOCK ? ROWS_PER_BLOCK : (int)rem64;
    if (remaining <= 0) return;

    // 64-bit global base of this tile (SGPR pair for GVS addressing).
    const unsigned long long tileBase =
        (unsigned long long)(const char*)x + (unsigned long long)tileRow0 * ROW_BYTES;
    // Low 32 bits of a generic LDS pointer == wave-relative LDS byte offset.
    const unsigned ldsBase = (unsigned)(size_t)smem;

    // Per-lane base addresses for chunk c = k*256 + t:
    //   global byte off = t*8 + k*2048, LDS byte off = (t>>3)*72 + (t&7)*8 + k*2304
    const unsigned goff0 = (unsigned)t * 8u;
    const unsigned ldso0 = ldsBase + ((unsigned)t >> 3) * (unsigned)LDS_ROW_BYTES
                                   + ((unsigned)t & 7u) * 8u;

    if (remaining == ROWS_PER_BLOCK) {
        // ---- Fast path: full tile, no EXEC-mask churn on the issue loop ----
#pragma unroll
        for (int k = 0; k < CHUNKS_PER_ROW; ++k) {
            asm volatile("global_load_async_to_lds_b64 %0, %1, %2"
                         :
                         : "v"(ldso0 + (unsigned)k * K_LSTRIDE),
                           "v"(goff0 + (unsigned)k * K_GSTRIDE),
                           "s"(tileBase)
                         : "memory");
        }
    } else {
        // ---- Tail path: guard each chunk against the ragged edge ----
        const unsigned validBytes = (unsigned)remaining * ROW_BYTES;
#pragma unroll
        for (int k = 0; k < CHUNKS_PER_ROW; ++k) {
            const unsigned goff = goff0 + (unsigned)k * K_GSTRIDE;
            if (goff < validBytes) {
                asm volatile("global_load_async_to_lds_b64 %0, %1, %2"
                             :
                             : "v"(ldso0 + (unsigned)k * K_LSTRIDE),
                               "v"(goff),
                               "s"(tileBase)
                             : "memory");
            }
        }
    }
    asm volatile("s_wait_asynccnt 0" ::: "memory");
    __syncthreads();

    if (t >= remaining) return;

    // ---- Conflict-free LDS reads of this lane's padded row ----
    const unsigned long long* rp = &smem[t * LDS_ROW_U64];
    unsigned long long m[8];
#pragma unroll
    for (int i = 0; i < 8; ++i) m[i] = rp[i];

    // ---- SipHash-2-4 with the standard test key ----
    const unsigned long long K0 = 0x0706050403020100ULL;
    const unsigned long long K1 = 0x0F0E0D0C0B0A0908ULL;
    unsigned long long v0 = 0x736F6D6570736575ULL ^ K0;
    unsigned long long v1 = 0x646F72616E646F6DULL ^ K1;
    unsigned long long v2 = 0x6C7967656E657261ULL ^ K0;
    unsigned long long v3 = 0x7465646279746573ULL ^ K1;

#pragma unroll
    for (int i = 0; i < 8; ++i) {
        v3 ^= m[i];
        sipround(v0, v1, v2, v3);
        sipround(v0, v1, v2, v3);
        v0 ^= m[i];
    }
    // Final padding word: total length (64 bytes) in the top byte.
    const unsigned long long fin = (unsigned long long)(ROW_BYTES & 0xFF) << 56;
    v3 ^= fin;
    sipround(v0, v1, v2, v3);
    sipround(v0, v1, v2, v3);
    v0 ^= fin;

    v2 ^= 0xFFULL;
    sipround(v0, v1, v2, v3);
    sipround(v0, v1, v2, v3);
    sipround(v0, v1, v2, v3);
    sipround(v0, v1, v2, v3);

    out[tileRow0 + t] = v0 ^ v1 ^ v2 ^ v3;
}

extern "C" void kernel_launch(void* const* d_in, const int* in_sizes, int n_in,
                              void* d_out, int out_size, void* d_ws, size_t ws_size,
                              hipStream_t stream) {
    (void)n_in; (void)d_ws; (void)ws_size; (void)out_size;
    const float* x = (const float*)d_in[0];
    unsigned long long* out = (unsigned long long*)d_out; // int64 hash bits
    const int rows = in_sizes[0] / 16;                    // 4,000,000
    const int grid = (rows + ROWS_PER_BLOCK - 1) / ROWS_PER_BLOCK;
    siphash_rows_kernel<<<grid, ROWS_PER_BLOCK, 0, stream>>>(x, out, rows);
}